// BSRPCell_31258771981066
// MI455X (gfx1250) — compile-verified
//
#include <hip/hip_runtime.h>
#include <hip/hip_bf16.h>
#include <math.h>

#define Nb   32
#define Ss   20
#define Dd   256
#define BEAM 4
#define MAXB 8
#define EPSF 1e-8f

typedef __attribute__((ext_vector_type(2))) float v2f;
typedef __attribute__((ext_vector_type(8))) float v8f;

__device__ __forceinline__ float gelu_f(float x) {
    // jax.nn.gelu approximate=False: x * 0.5 * (1 + erf(x / sqrt(2)))
    return 0.5f * x * (1.0f + erff(x * 0.7071067811865476f));
}
__device__ __forceinline__ float sigmoid_f(float x) {
    return 1.0f / (1.0f + expf(-x));
}

// ---------------------------------------------------------------------------
// C[M,Nn] = act(A[M,Kd] @ B[Kd,Nn] + bias), f32 WMMA 16x16x4.
// One wave (32 threads) per 16x16 output tile.
// A fragment (16x4): lanes 0-15 hold row M=l16 with K=kk+0/kk+1 in the two
// VGPRs; lanes 16-31 hold the same rows with K=kk+2/kk+3.
// B fragment (4x16): symmetric layout, N=l16 striped across lanes.
// C/D: VGPR i -> M = i (lanes 0-15) / M = 8+i (lanes 16-31), N = l16.
// ---------------------------------------------------------------------------
__global__ void wmma_gemm_kernel(const float* __restrict__ A,
                                 const float* __restrict__ B,
                                 const float* __restrict__ bias,
                                 float* __restrict__ C,
                                 int M, int Kd, int Nn, int act)
{
    int ntn  = Nn >> 4;
    int tile = blockIdx.x;
    int tm   = tile / ntn;
    int tn   = tile - tm * ntn;
    int lane = threadIdx.x;
    int half = lane >> 4;     // 0: K pair {0,1}, 1: K pair {2,3}
    int l16  = lane & 15;

    const float* Arow = A + (size_t)(tm * 16 + l16) * Kd;
    const float* Bcol = B + (size_t)(tn * 16 + l16);

    v8f c = {0.f, 0.f, 0.f, 0.f, 0.f, 0.f, 0.f, 0.f};
    for (int kk = 0; kk < Kd; kk += 4) {
        int ka = kk + half * 2;
        v2f a, b;
        a.x = Arow[ka];
        a.y = Arow[ka + 1];
        b.x = Bcol[(size_t)ka * Nn];
        b.y = Bcol[(size_t)(ka + 1) * Nn];
        c = __builtin_amdgcn_wmma_f32_16x16x4_f32(false, a, false, b,
                                                  (short)0, c, false, false);
    }

    int col  = tn * 16 + l16;
    float bv = bias[col];
    #pragma unroll
    for (int i = 0; i < 8; ++i) {
        int row = tm * 16 + half * 8 + i;
        float v = c[i] + bv;
        if (act == 1) v = gelu_f(v);
        C[(size_t)row * Nn + col] = v;
    }
}

// LayerNorm in place over rows of 256, eps=1e-5
__global__ void ln_kernel(float* __restrict__ x,
                          const float* __restrict__ g,
                          const float* __restrict__ b)
{
    __shared__ float s1[Dd];
    __shared__ float s2[Dd];
    int r = blockIdx.x;
    int d = threadIdx.x;
    float v = x[(size_t)r * Dd + d];
    s1[d] = v; s2[d] = v * v;
    __syncthreads();
    for (int off = Dd / 2; off > 0; off >>= 1) {
        if (d < off) { s1[d] += s1[d + off]; s2[d] += s2[d + off]; }
        __syncthreads();
    }
    float m   = s1[0] * (1.0f / Dd);
    float var = s2[0] * (1.0f / Dd) - m * m;
    x[(size_t)r * Dd + d] = (v - m) * rsqrtf(var + 1e-5f) * g[d] + b[d];
}

// stacks beam 0: rows 0,1 = seq[n,0/1], rest zero; last_mask one-hot blend.
__global__ void init_state_kernel(const float* __restrict__ seq,
                                  const float* __restrict__ input_mask,
                                  float* __restrict__ stacksA,
                                  float* __restrict__ lmaskA,
                                  int* __restrict__ posA)
{
    int n = blockIdx.x;
    int d = threadIdx.x;
    float* st = stacksA + (size_t)n * MAXB * Ss * Dd;  // beam 0
    for (int s = 0; s < Ss; ++s)
        st[s * Dd + d] = (s < 2) ? seq[((size_t)n * Ss + s) * Dd + d] : 0.0f;
    if (d < Ss) {
        float im = input_mask[n * Ss + 1];
        float v = 0.0f;
        if (d == 0) v = 1.0f - im;
        if (d == 1) v = im;
        lmaskA[(size_t)n * MAXB * Ss + d] = v;
    }
    if (d == 0) posA[n * MAXB] = 0;
}

// Per (n,k): last/laster soft reads, START blend, build win(768) / cc(512).
__global__ void prep_kernel(const float* __restrict__ stacksA,
                            const float* __restrict__ lmaskA,
                            const int* __restrict__ posA,
                            const float* __restrict__ seq,
                            const float* __restrict__ input_mask,
                            const float* __restrict__ start,
                            float* __restrict__ win,
                            float* __restrict__ cc,
                            float* __restrict__ last_items,
                            float* __restrict__ laster_items,
                            float* __restrict__ lp_arr,
                            float* __restrict__ cm_arr,
                            int K)
{
    int row = blockIdx.x;            // nk = n*K + k
    int n = row / K, k = row - n * K;
    int d = threadIdx.x;

    const float* st = stacksA + ((size_t)n * MAXB + k) * Ss * Dd;
    const float* lm = lmaskA  + ((size_t)n * MAXB + k) * Ss;

    float li = 0.f, lsi = 0.f, lp = 0.f;
    for (int s = 0; s < Ss; ++s) {
        float m  = lm[s];
        float ms = (s + 1 < Ss) ? lm[s + 1] : 0.0f;  // laster_mask[s]
        float v  = st[s * Dd + d];
        li  += m  * v;
        lsi += ms * v;
        if (s >= 1) lp += lm[s];                     // laster_prob
    }
    float lsi_f = lp * lsi + (1.0f - lp) * start[d];

    int p = posA[n * MAXB + k];
    float ci = 0.0f, cm = 0.0f;
    if (p + 2 < Ss) {
        ci = seq[((size_t)n * Ss + p + 2) * Dd + d];
        cm = input_mask[n * Ss + p + 2];
    }

    win[(size_t)row * (3 * Dd) + d]          = lsi_f;
    win[(size_t)row * (3 * Dd) + Dd + d]     = li;
    win[(size_t)row * (3 * Dd) + 2 * Dd + d] = ci;
    cc[(size_t)row * (2 * Dd) + d]           = lsi_f;
    cc[(size_t)row * (2 * Dd) + Dd + d]      = li;
    last_items[(size_t)row * Dd + d]   = li;
    laster_items[(size_t)row * Dd + d] = lsi_f;
    if (d == 0) { lp_arr[row] = lp; cm_arr[row] = cm; }
}

// dec = sigmoid(h . scorer_w + scorer_b); rs/ss log-prob math -> scoresB
__global__ void score_kernel(const float* __restrict__ h,
                             const float* __restrict__ scorer_w,
                             const float* __restrict__ scorer_b,
                             const float* __restrict__ lp_arr,
                             const float* __restrict__ cm_arr,
                             float* __restrict__ scoresB,
                             int K)
{
    __shared__ float red[Dd];
    int row = blockIdx.x;
    int d = threadIdx.x;
    red[d] = h[(size_t)row * Dd + d] * scorer_w[d];
    __syncthreads();
    for (int off = Dd / 2; off > 0; off >>= 1) {
        if (d < off) red[d] += red[d + off];
        __syncthreads();
    }
    if (d == 0) {
        float dec = sigmoid_f(red[0] + scorer_b[0]);
        float lp = lp_arr[row], cmf = cm_arr[row];
        float rs = dec, ss = 1.0f - dec;
        float both = ((lp == 0.0f) && (cmf == 0.0f)) ? 1.0f : 0.0f;
        rs = cmf * rs + (1.0f - cmf);
        rs = lp * rs;
        rs = both + (1.0f - both) * rs;
        rs = logf(rs + EPSF);
        ss = lp * ss + (1.0f - lp);
        ss = cmf * ss;
        ss = (1.0f - both) * ss;
        ss = logf(ss + EPSF);
        int n = row / K, k = row - (row / K) * K;
        scoresB[n * MAXB + k]     = rs;   // reduce beams 0..K-1
        scoresB[n * MAXB + K + k] = ss;   // shift beams  K..2K-1
    }
}

// gates + LN2 compose, then write red beam (k) and shift beam (K+k) into B.
__global__ void reduce_update_kernel(const float* __restrict__ contents,
                                     const float* __restrict__ laster_items,
                                     const float* __restrict__ last_items,
                                     const float* __restrict__ lp_arr,
                                     const float* __restrict__ cm_arr,
                                     const float* __restrict__ ln2_g,
                                     const float* __restrict__ ln2_b,
                                     const float* __restrict__ stacksA,
                                     const float* __restrict__ lmaskA,
                                     const int* __restrict__ posA,
                                     const float* __restrict__ seq,
                                     float* __restrict__ stacksB,
                                     float* __restrict__ lmaskB,
                                     int* __restrict__ posB,
                                     int K)
{
    __shared__ float s1[Dd];
    __shared__ float s2[Dd];
    int row = blockIdx.x;
    int n = row / K, k = row - n * K;
    int d = threadIdx.x;

    const float* crow = contents + (size_t)row * (4 * Dd);
    float g0 = sigmoid_f(crow[d]);
    float g1 = sigmoid_f(crow[Dd + d]);
    float g2 = sigmoid_f(crow[2 * Dd + d]);
    float parent = crow[3 * Dd + d];
    float lsi = laster_items[(size_t)row * Dd + d];
    float li  = last_items[(size_t)row * Dd + d];
    float x = g0 * lsi + g1 * li + g2 * parent;

    s1[d] = x; s2[d] = x * x;
    __syncthreads();
    for (int off = Dd / 2; off > 0; off >>= 1) {
        if (d < off) { s1[d] += s1[d + off]; s2[d] += s2[d + off]; }
        __syncthreads();
    }
    float m   = s1[0] * (1.0f / Dd);
    float var = s2[0] * (1.0f / Dd) - m * m;
    float ri  = (x - m) * rsqrtf(var + 1e-5f) * ln2_g[d] + ln2_b[d];

    float lp = lp_arr[row], cm = cm_arr[row];
    int p = posA[n * MAXB + k];
    float ci = (p + 2 < Ss) ? seq[((size_t)n * Ss + p + 2) * Dd + d] : 0.0f;

    const float* st = stacksA + ((size_t)n * MAXB + k) * Ss * Dd;
    const float* lm = lmaskA  + ((size_t)n * MAXB + k) * Ss;
    float* stR = stacksB + ((size_t)n * MAXB + k)     * Ss * Dd;
    float* stS = stacksB + ((size_t)n * MAXB + K + k) * Ss * Dd;

    for (int s = 0; s < Ss; ++s) {
        float lmv = lm[s];
        float lsm = (s + 1 < Ss) ? lm[s + 1] : 0.0f;   // laster_mask
        float sv  = st[s * Dd + d];
        // reduce branch: overwrite the laster slot with composed item
        stR[s * Dd + d] = lsm * ri + (1.0f - lsm) * sv;
        // shift branch
        float shm = (s > 0) ? lm[s - 1] : 0.0f;
        float shl = cm * shm + (1.0f - cm) * lmv;      // sh_lm after cm mix
        float tmp = shl * ci + (1.0f - shl) * sv;
        stS[s * Dd + d] = cm * tmp + (1.0f - cm) * sv;
    }
    if (d < Ss) {
        int s = d;
        float lmv = lm[s];
        float lsm = (s + 1 < Ss) ? lm[s + 1] : 0.0f;
        lmaskB[((size_t)n * MAXB + k) * Ss + s] = lp * lsm + (1.0f - lp) * lmv;
        float shm = (s > 0) ? lm[s - 1] : 0.0f;
        lmaskB[((size_t)n * MAXB + K + k) * Ss + s] = cm * shm + (1.0f - cm) * lmv;
    }
    if (d == 0) {
        posB[n * MAXB + k]     = p;
        posB[n * MAXB + K + k] = p + 1;
    }
}

// Stable descending top-k (jax.lax.top_k semantics: ties -> lower index).
// If twoK <= BEAM, identity (no selection in the reference).
__global__ void topk_kernel(const float* __restrict__ scoresB,
                            float* __restrict__ scoresA,
                            int* __restrict__ idx,
                            int twoK, int newK)
{
    int n = threadIdx.x;
    if (n >= Nb) return;
    const float* s = scoresB + n * MAXB;
    if (twoK <= BEAM) {
        for (int j = 0; j < newK; ++j) {
            idx[n * BEAM + j] = j;
            scoresA[n * BEAM + j] = s[j];
        }
        return;
    }
    bool used[MAXB];
    for (int i = 0; i < MAXB; ++i) used[i] = false;
    for (int j = 0; j < newK; ++j) {
        int best = 0;
        float bv = -3.402823466e38f;
        for (int i = 0; i < twoK; ++i) {
            float v = s[i];
            if (!used[i] && v > bv) { bv = v; best = i; }
        }
        used[best] = true;
        idx[n * BEAM + j] = best;
        scoresA[n * BEAM + j] = bv;
    }
}

__global__ void gather_kernel(const float* __restrict__ stacksB,
                              const float* __restrict__ lmaskB,
                              const int* __restrict__ posB,
                              const int* __restrict__ idx,
                              float* __restrict__ stacksA,
                              float* __restrict__ lmaskA,
                              int* __restrict__ posA,
                              int newK)
{
    int nk = blockIdx.x;
    int n = nk / newK, k = nk - n * newK;
    int src = idx[n * BEAM + k];
    int d = threadIdx.x;
    const float* sb = stacksB + ((size_t)n * MAXB + src) * Ss * Dd;
    float* sa       = stacksA + ((size_t)n * MAXB + k)   * Ss * Dd;
    for (int s = 0; s < Ss; ++s) sa[s * Dd + d] = sb[s * Dd + d];
    if (d < Ss)
        lmaskA[((size_t)n * MAXB + k) * Ss + d] =
            lmaskB[((size_t)n * MAXB + src) * Ss + d];
    if (d == 0) posA[n * MAXB + k] = posB[n * MAXB + src];
}

// out[n,d] = sum_k softmax(scores)[k] * (last_mask . stacks)[k,d]
__global__ void final_kernel(const float* __restrict__ stacksA,
                             const float* __restrict__ lmaskA,
                             const float* __restrict__ scoresA,
                             float* __restrict__ out)
{
    int n = blockIdx.x;
    int d = threadIdx.x;
    float mx = -3.402823466e38f;
    for (int k = 0; k < BEAM; ++k) mx = fmaxf(mx, scoresA[n * BEAM + k]);
    float p[BEAM], ssum = 0.0f;
    for (int k = 0; k < BEAM; ++k) { p[k] = expf(scoresA[n * BEAM + k] - mx); ssum += p[k]; }
    float inv = 1.0f / ssum;
    float acc = 0.0f;
    for (int k = 0; k < BEAM; ++k) {
        const float* st = stacksA + ((size_t)n * MAXB + k) * Ss * Dd;
        const float* lm = lmaskA  + ((size_t)n * MAXB + k) * Ss;
        float li = 0.0f;
        for (int s = 0; s < Ss; ++s) li += lm[s] * st[s * Dd + d];
        acc += p[k] * inv * li;
    }
    out[(size_t)n * Dd + d] = acc;
}

extern "C" void kernel_launch(void* const* d_in, const int* in_sizes, int n_in,
                              void* d_out, int out_size, void* d_ws, size_t ws_size,
                              hipStream_t stream)
{
    const float* sequence   = (const float*)d_in[0];
    const float* input_mask = (const float*)d_in[1];
    const float* w_init     = (const float*)d_in[2];
    const float* b_init     = (const float*)d_in[3];
    const float* ln1_g      = (const float*)d_in[4];
    const float* ln1_b      = (const float*)d_in[5];
    const float* scorer_w   = (const float*)d_in[6];
    const float* scorer_b   = (const float*)d_in[7];
    const float* conv_w     = (const float*)d_in[8];
    const float* conv_b     = (const float*)d_in[9];
    const float* start      = (const float*)d_in[10];
    const float* wcell1_w   = (const float*)d_in[11];
    const float* wcell1_b   = (const float*)d_in[12];
    const float* wcell2_w   = (const float*)d_in[13];
    const float* wcell2_b   = (const float*)d_in[14];
    const float* ln2_g      = (const float*)d_in[15];
    const float* ln2_b      = (const float*)d_in[16];
    float* out = (float*)d_out;

    char* wp = (char*)d_ws;
    auto alloc = [&](size_t bytes) -> void* {
        void* p = (void*)wp;
        wp += (bytes + 255) & ~(size_t)255;
        return p;
    };
    const size_t Mmax = (size_t)Nb * BEAM;  // 128 rows max through the GEMMs
    float* seq          = (float*)alloc((size_t)Nb * Ss * Dd * 4);
    float* stacksA      = (float*)alloc((size_t)Nb * MAXB * Ss * Dd * 4);
    float* stacksB      = (float*)alloc((size_t)Nb * MAXB * Ss * Dd * 4);
    float* lmaskA       = (float*)alloc((size_t)Nb * MAXB * Ss * 4);
    float* lmaskB       = (float*)alloc((size_t)Nb * MAXB * Ss * 4);
    int*   posA         = (int*)  alloc((size_t)Nb * MAXB * 4);
    int*   posB         = (int*)  alloc((size_t)Nb * MAXB * 4);
    float* scoresA      = (float*)alloc((size_t)Nb * BEAM * 4);
    float* scoresB      = (float*)alloc((size_t)Nb * MAXB * 4);
    int*   idxBuf       = (int*)  alloc((size_t)Nb * BEAM * 4);
    float* winBuf       = (float*)alloc(Mmax * 3 * Dd * 4);
    float* ccBuf        = (float*)alloc(Mmax * 2 * Dd * 4);
    float* hBuf         = (float*)alloc(Mmax * Dd * 4);
    float* interBuf     = (float*)alloc(Mmax * 4 * Dd * 4);
    float* contentsBuf  = (float*)alloc(Mmax * 4 * Dd * 4);
    float* lastBuf      = (float*)alloc(Mmax * Dd * 4);
    float* lasterBuf    = (float*)alloc(Mmax * Dd * 4);
    float* lpBuf        = (float*)alloc(Mmax * 4);
    float* cmBuf        = (float*)alloc(Mmax * 4);
    (void)ws_size; (void)in_sizes; (void)n_in; (void)out_size;

    auto gemm = [&](const float* A, const float* B, const float* bias, float* C,
                    int M, int Kd, int Nn, int act) {
        int tiles = (M / 16) * (Nn / 16);
        wmma_gemm_kernel<<<tiles, 32, 0, stream>>>(A, B, bias, C, M, Kd, Nn, act);
    };

    // seq = LN1(sequence @ w_init + b_init)
    gemm(sequence, w_init, b_init, seq, Nb * Ss, Dd, Dd, 0);
    ln_kernel<<<Nb * Ss, Dd, 0, stream>>>(seq, ln1_g, ln1_b);
    init_state_kernel<<<Nb, Dd, 0, stream>>>(seq, input_mask, stacksA, lmaskA, posA);

    int K = 1;
    for (int t = 2; t < 2 * Ss - 1; ++t) {   // 37 iterations
        int M = Nb * K;
        prep_kernel<<<M, Dd, 0, stream>>>(stacksA, lmaskA, posA, seq, input_mask,
                                          start, winBuf, ccBuf, lastBuf, lasterBuf,
                                          lpBuf, cmBuf, K);
        gemm(winBuf, conv_w, conv_b, hBuf, M, 3 * Dd, Dd, 1);          // gelu
        score_kernel<<<M, Dd, 0, stream>>>(hBuf, scorer_w, scorer_b,
                                           lpBuf, cmBuf, scoresB, K);
        gemm(ccBuf, wcell1_w, wcell1_b, interBuf, M, 2 * Dd, 4 * Dd, 1); // gelu
        gemm(interBuf, wcell2_w, wcell2_b, contentsBuf, M, 4 * Dd, 4 * Dd, 0);
        reduce_update_kernel<<<M, Dd, 0, stream>>>(contentsBuf, lasterBuf, lastBuf,
                                                   lpBuf, cmBuf, ln2_g, ln2_b,
                                                   stacksA, lmaskA, posA, seq,
                                                   stacksB, lmaskB, posB, K);
        int twoK = 2 * K;
        int newK = (twoK > BEAM) ? BEAM : twoK;
        topk_kernel<<<1, Nb, 0, stream>>>(scoresB, scoresA, idxBuf, twoK, newK);
        gather_kernel<<<Nb * newK, Dd, 0, stream>>>(stacksB, lmaskB, posB, idxBuf,
                                                    stacksA, lmaskA, posA, newK);
        K = newK;
    }

    final_kernel<<<Nb, Dd, 0, stream>>>(stacksA, lmaskA, scoresA, out);
}